// model_benchmark_concat_marabou_58548994179557
// MI455X (gfx1250) — compile-verified
//
#include <hip/hip_runtime.h>

typedef __attribute__((ext_vector_type(2))) float v2f;
typedef __attribute__((ext_vector_type(4))) float v4f;
typedef __attribute__((ext_vector_type(8))) float v8f;

// ---------------- LDS layout (floats) ----------------
// Weights padded to [Kp x Np] (Np >= 16 so B-fragment col reads are in-range,
// K-pad rows zeroed so they contribute 0 to every output).
static constexpr int g_Woff[16] = {   0, 128, 384,  512, 640, 896,
                                   1024,1280,1536, 1664,1792,2048,
                                   2176,3200,3712, 3968};
static constexpr int g_K  [16] = { 5,16, 8,  8,16, 8, 13,16, 8,  8,16, 8, 29,32,16, 8};
static constexpr int g_Kp [16] = { 8,16, 8,  8,16, 8, 16,16, 8,  8,16, 8, 32,32,16, 8};
static constexpr int g_N  [16] = {16, 8, 8, 16, 8, 8, 16, 8, 8, 16, 8, 8, 32,16, 8, 1};
static constexpr int g_Np [16] = {16,16,16, 16,16,16, 16,16,16, 16,16,16, 32,16,16,16};

#define BIAS_OFF   4096   // 16 biases x 32 floats
#define GCN_OFF    4608   // 8 x 400
#define SUMM_OFF   7808   // 400
#define BMAP_OFF   8208   // 400
#define RUN_OFF    8608   // 32
#define WS_OFF     8640   // per-wave workspace
#define WS_STRIDE  2736
#define LDS_TOTAL  (WS_OFF + 2*WS_STRIDE)   // 14112 floats = 56448 B < 64KB

struct KParams {
  const float* inp;
  const float* w[16];
  const float* b[16];
  const float* gcn;
  const float* summ;
  const float* run;
  const float* bmap;
  float* out;
};

// D[20xN] = relu?( A[20xKP] @ W[KP x N] + bias ), fully constant-folded geometry.
// A rows >=20 and W cols >= N may be garbage: they only feed masked-out outputs.
// K-padding (cols of A / rows of W beyond the true K) MUST be zero.
template<int LDA, int LDW, int LDO, int KP, int N, bool RELU, bool HASBIAS>
__device__ __forceinline__ void gemm(const float* A, const float* W,
                                     const float* bias, float* O, int lane) {
  constexpr int NT = (N + 15) / 16;
  const int half = lane >> 4;
  const int lm   = lane & 15;
#pragma unroll
  for (int nt = 0; nt < NT; ++nt) {
    const int nb  = nt * 16;
    const int col = nb + lm;
    const float bv = HASBIAS ? bias[nb + lm] : 0.0f;
    // col-mask folds to 'true' for full 16-wide tiles
    const bool colok = (N - nb >= 16) ? true : (col < N);
    const float* wp = W + (2 * half) * LDW + col;      // W[k+2*half (+1)][col]

    // ---- M-tile 0: rows 0..15, never masked by row ----
    {
      v8f acc = {bv, bv, bv, bv, bv, bv, bv, bv};
      const float* ap = A + lm * LDA + 2 * half;       // A[row][k+2*half (+1)]
#pragma unroll
      for (int k = 0; k < KP; k += 4) {
        v2f a; a.x = ap[k];       a.y = ap[k + 1];
        v2f b; b.x = wp[k * LDW]; b.y = wp[k * LDW + LDW];
        acc = __builtin_amdgcn_wmma_f32_16x16x4_f32(false, a, false, b,
                                                    (short)0, acc, false, false);
      }
      if (colok) {
#pragma unroll
        for (int i = 0; i < 8; ++i) {                  // row = i + 8*half in 0..15
          const float v = acc[i];
          O[(i + 8 * half) * LDO + col] = RELU ? fmaxf(v, 0.0f) : v;
        }
      }
    }
    // ---- M-tile 1: rows 16..31, only rows 16..19 (half==0, i<4) are real ----
    {
      v8f acc = {bv, bv, bv, bv, bv, bv, bv, bv};
      const float* ap = A + (16 + lm) * LDA + 2 * half;
#pragma unroll
      for (int k = 0; k < KP; k += 4) {
        v2f a; a.x = ap[k];       a.y = ap[k + 1];
        v2f b; b.x = wp[k * LDW]; b.y = wp[k * LDW + LDW];
        acc = __builtin_amdgcn_wmma_f32_16x16x4_f32(false, a, false, b,
                                                    (short)0, acc, false, false);
      }
      if (colok && half == 0) {
#pragma unroll
        for (int i = 0; i < 4; ++i) {                  // rows 16..19
          const float v = acc[i];
          O[(16 + i) * LDO + col] = RELU ? fmaxf(v, 0.0f) : v;
        }
      }
    }
  }
}

template<int LI, int LDA, int LDO, bool RELU>
__device__ __forceinline__ void lin(const float* A, float* O, float* lds, int lane) {
  gemm<LDA, g_Np[LI], LDO, g_Kp[LI], g_N[LI], RELU, true>(
      A, lds + g_Woff[LI], lds + BIAS_OFF + LI * 32, O, lane);
}

// 3-layer block: in -> T16 -> T8 -> out (all ReLU)
template<int BASE, int LDIN, int LDO>
__device__ __forceinline__ void block3(const float* in, float* t16, float* t8,
                                       float* out, float* lds, int lane) {
  lin<BASE + 0, LDIN, 16, true>(in,  t16, lds, lane);
  lin<BASE + 1, 16,   8,  true>(t16, t8,  lds, lane);
  lin<BASE + 2, 8,    LDO, true>(t8, out, lds, lane);
}

__global__ __launch_bounds__(64)
void gnn_fused_kernel(KParams P) {
  __shared__ alignas(16) float lds[LDS_TOTAL];
  const int tid  = threadIdx.x;
  const int lane = tid & 31;
  const int wave = tid >> 5;

  // Zero all LDS with b128 stores (establishes every K-pad as exactly 0).
  {
    v4f* z = (v4f*)lds;
    const v4f zero = {0.f, 0.f, 0.f, 0.f};
    for (int i = tid; i < LDS_TOTAL / 4; i += 64) z[i] = zero;
  }
  __syncthreads();

  // Stage padded weights + biases (scalar: layouts are re-strided) ...
#pragma unroll
  for (int w = 0; w < 16; ++w) {
    const float* src = P.w[w];
    const int K = g_K[w], N = g_N[w], Np = g_Np[w];
    float* dst = lds + g_Woff[w];
    for (int i = tid; i < K * N; i += 64) dst[(i / N) * Np + (i % N)] = src[i];
    if (tid < N) lds[BIAS_OFF + w * 32 + tid] = P.b[w][tid];
  }
  // ... and contiguous graph matrices with b128 copies.
  {
    const v4f* gs = (const v4f*)P.gcn;
    v4f* gd = (v4f*)(lds + GCN_OFF);
    for (int i = tid; i < 800; i += 64) gd[i] = gs[i];
    const v4f* ss = (const v4f*)P.summ;
    v4f* sd = (v4f*)(lds + SUMM_OFF);
    for (int i = tid; i < 100; i += 64) sd[i] = ss[i];
    const v4f* bs = (const v4f*)P.bmap;
    v4f* bd = (v4f*)(lds + BMAP_OFF);
    for (int i = tid; i < 100; i += 64) bd[i] = bs[i];
  }
  if (tid < 20) lds[RUN_OFF + tid] = P.run[tid];
  __syncthreads();

  // ---------- per-wave independent pass (wave 0: original, wave 1: perturbed) ----------
  float* ws    = lds + WS_OFF + wave * WS_STRIDE;
  float* NI    = ws + 0;     // 20x8  (cols 5..7 zero)
  float* X     = ws + 160;   // 20x8
  float* Y     = ws + 320;   // 20x8
  float* G     = ws + 480;   // 20x8
  float* T16   = ws + 640;   // 20x16
  float* T8    = ws + 960;   // 20x8
  float* S13   = ws + 1120;  // 20x16 (cols 13..15 zero); reused as fc4 output
  float* MERGE = ws + 1440;  // 20x32 (cols 29..31 zero)
  float* F1    = ws + 2080;  // 20x32
  float* GG    = ws + 2720;  // 16

  for (int i = lane; i < 100; i += 32) NI[(i / 5) * 8 + (i % 5)] = P.inp[i];
  if (wave == 1 && lane == 0) { NI[0] += 1.0f / 15.0f; NI[2] -= 1.0f / 15.0f; }

  // h block: [20,5] -> [20,8]
  block3<0, 8, 8>(NI, T16, T8, X, lds, lane);
  // y = f(x)
  block3<3, 8, 8>(X, T16, T8, Y, lds, lane);
  // 8-step message passing
  for (int it = 0; it < 8; ++it) {
    gemm<20, 8, 8, 20, 8, false, false>(lds + GCN_OFF + it * 400, Y, nullptr, G, lane);
    block3<3, 8, 8>(G, T16, T8, Y, lds, lane);
    for (int i = lane; i < 160; i += 32) Y[i] += X[i];
    if (it < 7) block3<3, 8, 8>(Y, T16, T8, Y, lds, lane);
  }
  // s = concat([node_inputs, gcn_output]) -> [20,13] (padded 16)
  for (int i = lane; i < 20 * 13; i += 32) {
    const int r = i / 13, c = i % 13;
    S13[r * 16 + c] = (c < 5) ? NI[r * 8 + c] : Y[r * 8 + (c - 5)];
  }
  // dag block -> X
  block3<6, 16, 8>(S13, T16, T8, X, lds, lane);
  // gsn_dag_summary = summ_mats @ s  -> G
  gemm<20, 8, 8, 20, 8, false, false>(lds + SUMM_OFF, X, nullptr, G, lane);
  // dag_summ_extend = backward_map @ gsn_dag_summary -> X (before G is overwritten)
  gemm<20, 8, 8, 20, 8, false, false>(lds + BMAP_OFF, G, nullptr, X, lane);
  // g block (global_gc): G -> G
  block3<9, 8, 8>(G, T16, T8, G, lds, lane);
  // gsn_global_summary = running_dags_mat @ G  -> GG[8]
  if (lane < 8) {
    float s = 0.0f;
#pragma unroll
    for (int k = 0; k < 20; ++k) s += lds[RUN_OFF + k] * G[k * 8 + lane];
    GG[lane] = s;
  }
  // merge = concat([NI(5) | gcn_out(8) | dag_extend(8) | global_extend(8)]) -> [20,29] pad 32
  for (int i = lane; i < 20 * 29; i += 32) {
    const int r = i / 29, c = i % 29;
    float v;
    if      (c <  5) v = NI[r * 8 + c];
    else if (c < 13) v = Y [r * 8 + c - 5];
    else if (c < 21) v = X [r * 8 + c - 13];
    else             v = GG[c - 21];
    MERGE[r * 32 + c] = v;
  }
  // fc head
  lin<12, 32, 32, true >(MERGE, F1,  lds, lane);
  lin<13, 32, 16, true >(F1,    T16, lds, lane);
  lin<14, 16, 8,  true >(T16,   T8,  lds, lane);
  lin<15, 8,  16, false>(T8,    S13, lds, lane);   // fc4, N=1, no ReLU
  // node_outputs + node_valid_mask * 10000
  if (lane < 20)
    P.out[wave * 20 + lane] = S13[lane * 16] + P.inp[100 + lane] * 10000.0f;
}

extern "C" void kernel_launch(void* const* d_in, const int* in_sizes, int n_in,
                              void* d_out, int out_size, void* d_ws, size_t ws_size,
                              hipStream_t stream) {
  (void)d_ws; (void)ws_size; (void)out_size;
  KParams P;
  P.inp = (const float*)d_in[0];
  P.out = (float*)d_out;

  if (n_in >= 37 && in_sizes[1] == 80) {
    // params flattened in dict insertion order: h1_w,h1_b,...,fc4_b,gcn,summ,run,bmap
    int idx = 1;
    for (int w = 0; w < 16; ++w) {
      P.w[w] = (const float*)d_in[idx++];
      P.b[w] = (const float*)d_in[idx++];
    }
    P.gcn  = (const float*)d_in[33];
    P.summ = (const float*)d_in[34];
    P.run  = (const float*)d_in[35];
    P.bmap = (const float*)d_in[36];
  } else if (n_in >= 37) {
    // jax-pytree sorted-key order
    const int wi[16] = {30,32,34, 9,11,13, 2,4,6, 23,25,27, 15,17,19,21};
    const int bi[16] = {29,31,33, 8,10,12, 1,3,5, 22,24,26, 14,16,18,20};
    for (int w = 0; w < 16; ++w) {
      P.w[w] = (const float*)d_in[wi[w]];
      P.b[w] = (const float*)d_in[bi[w]];
    }
    P.gcn  = (const float*)d_in[28];
    P.summ = (const float*)d_in[36];
    P.run  = (const float*)d_in[35];
    P.bmap = (const float*)d_in[7];
  } else {
    // params passed as one concatenated blob (insertion order offsets)
    const float* blob = (const float*)d_in[1];
    const int woff[16] = {   0,  96, 232, 304, 448, 584,  656,  880, 1016,
                          1088,1232,1368, 1440,2400,2928, 3064};
    const int boff[16] = {  80, 224, 296, 432, 576, 648,  864, 1008, 1080,
                          1216,1360,1432, 2368,2912,3056, 3072};
    for (int w = 0; w < 16; ++w) { P.w[w] = blob + woff[w]; P.b[w] = blob + boff[w]; }
    P.gcn  = blob + 3073;
    P.summ = blob + 6273;
    P.run  = blob + 6673;
    P.bmap = blob + 6693;
  }

  hipLaunchKernelGGL(gnn_fused_kernel, dim3(1), dim3(64), 0, stream, P);
}